// qnn_graph_v7_81681688035671
// MI455X (gfx1250) — compile-verified
//
#include <hip/hip_runtime.h>

typedef float v2f __attribute__((ext_vector_type(2)));
typedef float v8f __attribute__((ext_vector_type(8)));

#define Tt 4096
#define Hh 32
#define Nn 64
#define Cc 16
#define NCHUNK (Tt / Cc)
#define ST 68    // row stride (floats) for 16-row LDS matrices (pad vs 64 banks)
#define SSTR 68  // row stride for 64x64 state in LDS

__device__ __forceinline__ v8f wmma4(v2f a, v2f b, v8f c) {
  // D(16x16,f32) = A(16x4,f32) * B(4x16,f32) + C
  return __builtin_amdgcn_wmma_f32_16x16x4_f32(false, a, false, b, (short)0, c, false, false);
}

// Accumulate C += A(16xK) * B(Kx16).
// A read row-major from LDS: A(m,k) = A[(ar0+m)*As + k]
// BT=false: B(k,n) = B[k*Bs + bn0+n]   (row-major, rows = K)
// BT=true : B(k,n) = B[(bn0+n)*Bs + k] (transposed read)
template <bool BT, int K>
__device__ __forceinline__ v8f mmA(const float* A, int As, int ar0,
                                   const float* B, int Bs, int bn0,
                                   v8f c, int mr, int hb) {
#pragma unroll
  for (int k0 = 0; k0 < K; k0 += 4) {
    int kk = k0 + hb;  // this lane-half covers K offsets {kk, kk+1}
    v2f av, bv;
    const float* pa = A + (ar0 + mr) * As + kk;
    av.x = pa[0];
    av.y = pa[1];
    if (!BT) {
      bv.x = B[kk * Bs + bn0 + mr];
      bv.y = B[(kk + 1) * Bs + bn0 + mr];
    } else {
      const float* pb = B + (bn0 + mr) * Bs + kk;
      bv.x = pb[0];
      bv.y = pb[1];
    }
    c = wmma4(av, bv, c);
  }
  return c;
}

__global__ __launch_bounds__(128) void rwkv7_wkv_chunked(
    const float* __restrict__ r, const float* __restrict__ w,
    const float* __restrict__ k, const float* __restrict__ v,
    const float* __restrict__ a, const float* __restrict__ b,
    const float* __restrict__ s0, float* __restrict__ y,
    float* __restrict__ sOut) {
  __shared__ float sS[Nn * SSTR];               // state, rows=i (value), cols=j (key)
  __shared__ float sAh[Cc * ST], sRh[Cc * ST];  // a*W_{t-1}, r*W_t
  __shared__ float sBd[Cc * ST], sKd[Cc * ST];  // b/W_t, k/W_t
  __shared__ float sV[Cc * ST], sU[Cc * ST];    // raw v, solved U (u_t = S_{t-1} a_t)
  __shared__ float sM[4][16 * 17];              // Mab, Mak, Srb, Srk (C x C)
  __shared__ float sEL[Nn];                     // W_C per column

  const int h = blockIdx.x;
  const int tid = threadIdx.x;
  const int lane = tid & 31;
  const int wv_ = tid >> 5;        // wave id 0..3
  const int mr = lane & 15;        // M-row / N-col index within tile
  const int hb = (lane >> 4) * 2;  // K sub-offset for A/B operands
  const int mh = (lane >> 4) * 8;  // M offset for C/D tiles
  const int i0 = wv_ * 16;         // this wave's state-row slice

  // Per-thread register double buffer for chunk inputs (software pipeline).
  float pr[Cc], pw[Cc], pk[Cc], pa[Cc], pb[Cc];  // threads 0..63 (column j = tid)
  float pv[Cc];                                  // threads 64..127 (column j = tid-64)

  // load initial state
  for (int idx = tid; idx < Nn * Nn; idx += 128)
    sS[(idx >> 6) * SSTR + (idx & 63)] = s0[h * Nn * Nn + idx];

  // prefetch chunk 0
  if (tid < 64) {
    const int base = h * Nn + tid;  // t0 = 0
#pragma unroll
    for (int t = 0; t < Cc; ++t) {
      const int g = base + t * Hh * Nn;
      pr[t] = r[g]; pw[t] = w[g]; pk[t] = k[g]; pa[t] = a[g]; pb[t] = b[g];
    }
  } else {
    const int base = h * Nn + (tid - 64);
#pragma unroll
    for (int t = 0; t < Cc; ++t) pv[t] = v[base + t * Hh * Nn];
  }
  __syncthreads();

  for (int chunk = 0; chunk < NCHUNK; ++chunk) {
    const int t0 = chunk * Cc;

    // ---- phase 1: elementwise prep from registers (decay products per column) ----
    if (tid < 64) {
      const int j = tid;
      float Wc = 1.0f;
#pragma unroll
      for (int t = 0; t < Cc; ++t) {
        sAh[t * ST + j] = pa[t] * Wc;  // W_{t-1}
        Wc *= pw[t];                   // W_t
        const float inv = 1.0f / Wc;
        sRh[t * ST + j] = pr[t] * Wc;
        sBd[t * ST + j] = pb[t] * inv;
        sKd[t * ST + j] = pk[t] * inv;
      }
      sEL[j] = Wc;  // W_C
    } else {
      const int j = tid - 64;
#pragma unroll
      for (int t = 0; t < Cc; ++t) sV[t * ST + j] = pv[t];
    }
    __syncthreads();

    // ---- prefetch next chunk's inputs (latency hidden by WMMA phases below) ----
    if (chunk + 1 < NCHUNK) {
      const int t0n = t0 + Cc;
      if (tid < 64) {
        const int base = (t0n * Hh + h) * Nn + tid;
#pragma unroll
        for (int t = 0; t < Cc; ++t) {
          const int g = base + t * Hh * Nn;
          pr[t] = r[g]; pw[t] = w[g]; pk[t] = k[g]; pa[t] = a[g]; pb[t] = b[g];
        }
      } else {
        const int base = (t0n * Hh + h) * Nn + (tid - 64);
#pragma unroll
        for (int t = 0; t < Cc; ++t) pv[t] = v[base + t * Hh * Nn];
      }
    }

    // ---- (a) C x C score matrices, one per wave ----
    {
      const float* Am = (wv_ < 2) ? sAh : sRh;
      const float* Bm = (wv_ & 1) ? sKd : sBd;
      v8f c = {0, 0, 0, 0, 0, 0, 0, 0};
      c = mmA<true, 64>(Am, ST, 0, Bm, ST, 0, c, mr, hb);  // (Ah|Rh) @ (Bd|Kd)^T
#pragma unroll
      for (int p = 0; p < 8; ++p) {
        const int t = p + mh;
        const bool keep = (wv_ < 2) ? (mr < t) : (mr <= t);  // strict vs inclusive lower
        if (!keep) c[p] = 0.0f;
      }
#pragma unroll
      for (int p = 0; p < 8; ++p) sM[wv_][(p + mh) * 17 + mr] = c[p];
    }
    __syncthreads();

    // ---- (b) RHS = Ah @ S0^T + Mak @ V  -> sU (this wave's 16-column slice) ----
    {
      v8f c = {0, 0, 0, 0, 0, 0, 0, 0};
      c = mmA<true, 64>(sAh, ST, 0, sS, SSTR, i0, c, mr, hb);   // Ah @ S0^T
      c = mmA<false, 16>(sM[1], 17, 0, sV, ST, i0, c, mr, hb);  // Mak @ V
#pragma unroll
      for (int p = 0; p < 8; ++p) sU[(p + mh) * ST + i0 + mr] = c[p];
    }
    __syncthreads();

    // ---- (c) forward substitution: U = RHS + Mab @ U (columns independent) ----
    if (tid < 64) {
      const int cc = tid;
      float ucol[16];
#pragma unroll
      for (int t = 0; t < 16; ++t) {
        float u = sU[t * ST + cc];
#pragma unroll
        for (int s = 0; s < 16; ++s)
          if (s < t) u += sM[0][t * 17 + s] * ucol[s];
        ucol[t] = u;
      }
#pragma unroll
      for (int t = 0; t < 16; ++t) sU[t * ST + cc] = ucol[t];
    }
    __syncthreads();

    // ---- (d) outputs: Y = Rh @ S0^T + Srb @ U + Srk @ V ----
    {
      v8f c = {0, 0, 0, 0, 0, 0, 0, 0};
      c = mmA<true, 64>(sRh, ST, 0, sS, SSTR, i0, c, mr, hb);
      c = mmA<false, 16>(sM[2], 17, 0, sU, ST, i0, c, mr, hb);
      c = mmA<false, 16>(sM[3], 17, 0, sV, ST, i0, c, mr, hb);
#pragma unroll
      for (int p = 0; p < 8; ++p) {
        const int t = p + mh;
        __builtin_nontemporal_store(c[p], &y[((t0 + t) * Hh + h) * Nn + i0 + mr]);
      }
    }

    // ---- (e) state update: S = S .* W_C + U^T @ (Bd .* W_C) + V^T @ (Kd .* W_C) ----
    {
#pragma unroll
      for (int tj = 0; tj < 4; ++tj) {
        const int j0 = tj * 16;
        const float esc = sEL[j0 + mr];  // W_C for this lane's column
        v8f s;
#pragma unroll
        for (int p = 0; p < 8; ++p) s[p] = sS[(i0 + p + mh) * SSTR + j0 + mr] * esc;
#pragma unroll
        for (int k0 = 0; k0 < 16; k0 += 4) {
          const int kk = k0 + hb;
          v2f av, bv;
          av.x = sU[kk * ST + i0 + mr];  // A = U^T  (m=i, k=s)
          av.y = sU[(kk + 1) * ST + i0 + mr];
          bv.x = sBd[kk * ST + j0 + mr] * esc;  // B = Bh (k=s, n=j)
          bv.y = sBd[(kk + 1) * ST + j0 + mr] * esc;
          s = wmma4(av, bv, s);
          av.x = sV[kk * ST + i0 + mr];  // A = V^T
          av.y = sV[(kk + 1) * ST + i0 + mr];
          bv.x = sKd[kk * ST + j0 + mr] * esc;  // B = Kh
          bv.y = sKd[(kk + 1) * ST + j0 + mr] * esc;
          s = wmma4(av, bv, s);
        }
#pragma unroll
        for (int p = 0; p < 8; ++p) sS[(i0 + p + mh) * SSTR + j0 + mr] = s[p];
      }
    }
    __syncthreads();
  }

  // final state out
  for (int idx = tid; idx < Nn * Nn; idx += 128)
    sOut[h * Nn * Nn + idx] = sS[(idx >> 6) * SSTR + (idx & 63)];
}

extern "C" void kernel_launch(void* const* d_in, const int* in_sizes, int n_in,
                              void* d_out, int out_size, void* d_ws, size_t ws_size,
                              hipStream_t stream) {
  const float* r = (const float*)d_in[0];
  const float* w = (const float*)d_in[1];
  const float* k = (const float*)d_in[2];
  const float* v = (const float*)d_in[3];
  const float* a = (const float*)d_in[4];
  const float* b = (const float*)d_in[5];
  const float* s0 = (const float*)d_in[6];
  float* y = (float*)d_out;                // [T,H,N]
  float* sOut = y + (size_t)Tt * Hh * Nn;  // [H,N,N]
  rwkv7_wkv_chunked<<<dim3(Hh), dim3(128), 0, stream>>>(r, w, k, v, a, b, s0, y, sOut);
}